// HeteroGCN_34385508171930
// MI455X (gfx1250) — compile-verified
//
#include <hip/hip_runtime.h>
#include <hip/hip_bf16.h>

typedef __attribute__((ext_vector_type(2))) float v2f;
typedef __attribute__((ext_vector_type(8))) float v8f;
typedef __attribute__((ext_vector_type(4))) int   v4i;

#define NG 100000
#define NT 100000
#define NE 500000
#define DD 128

// Detect the CDNA5 async-to-LDS builtin (toolchain-dependent); fall back to
// plain VGPR staging if absent so the file always compiles.
#define HGCN_ASYNC_LDS 0
#ifdef __HIP_DEVICE_COMPILE__
#ifdef __has_builtin
#if __has_builtin(__builtin_amdgcn_global_load_async_to_lds_b128)
#undef HGCN_ASYNC_LDS
#define HGCN_ASYNC_LDS 1
#endif
#endif
#endif

#if HGCN_ASYNC_LDS
#ifdef __has_builtin
#if __has_builtin(__builtin_amdgcn_s_wait_asynccnt)
#define HGCN_WAIT_ASYNC() __builtin_amdgcn_s_wait_asynccnt(0)
#endif
#endif
#ifndef HGCN_WAIT_ASYNC
#define HGCN_WAIT_ASYNC() asm volatile("s_wait_asynccnt 0" ::: "memory")
#endif
#endif

// ---------------------------------------------------------------------------
// Kernel 1: weight prep. Builds 5 transposed weight matrices in workspace so
// GEMM B-fragments become contiguous 8-byte loads:
//   WT[0] = (Ws_gg + Ws_tg)^T   WT[1] = Wn_gg^T   WT[2] = Wn_tg^T
//   WT[3] = Ws_gt^T             WT[4] = Wn_gt^T
// ---------------------------------------------------------------------------
__global__ void hgcn_wprep_kernel(const float* __restrict__ ws_gg,
                                  const float* __restrict__ ws_tg,
                                  const float* __restrict__ wn_gg,
                                  const float* __restrict__ wn_tg,
                                  const float* __restrict__ ws_gt,
                                  const float* __restrict__ wn_gt,
                                  float* __restrict__ wt) {
    int i = blockIdx.x * blockDim.x + threadIdx.x;   // k*128 + n
    if (i >= DD * DD) return;
    int k = i >> 7, n = i & (DD - 1);
    int o = n * DD + k;                              // transposed index
    wt[0 * DD * DD + o] = ws_gg[i] + ws_tg[i];
    wt[1 * DD * DD + o] = wn_gg[i];
    wt[2 * DD * DD + o] = wn_tg[i];
    wt[3 * DD * DD + o] = ws_gt[i];
    wt[4 * DD * DD + o] = wn_gt[i];
}

// ---------------------------------------------------------------------------
// Kernel 2: edge scatter. One wave per edge: coalesced float4 gather of the
// 128-float src row, f32 atomic accumulate into dst row + degree bump.
// ---------------------------------------------------------------------------
__global__ __launch_bounds__(256)
void hgcn_scatter_kernel(const float* __restrict__ xsrc,
                         const int* __restrict__ src,
                         const int* __restrict__ dst,
                         float* __restrict__ sum,
                         float* __restrict__ deg,
                         int nE) {
    int edge = (blockIdx.x * blockDim.x + threadIdx.x) >> 5;
    int lane = threadIdx.x & 31;
    if (edge >= nE) return;
    int s = src[edge];
    int d = dst[edge];
    const float4* srow = (const float4*)(xsrc + (size_t)s * DD);
    float4 v = srow[lane];                           // 32 lanes * 4 = 128
    float* drow = sum + (size_t)d * DD + lane * 4;
    atomicAdd(drow + 0, v.x);
    atomicAdd(drow + 1, v.y);
    atomicAdd(drow + 2, v.z);
    atomicAdd(drow + 3, v.w);
    if (lane == 0) atomicAdd(deg + d, 1.0f);
}

// ---------------------------------------------------------------------------
// Kernel 3: h_neigh = sum / max(deg, 1), float4-vectorized.
// ---------------------------------------------------------------------------
__global__ __launch_bounds__(256)
void hgcn_mean_kernel(float4* __restrict__ sum4,
                      const float* __restrict__ deg,
                      int nQuads) {
    int i = blockIdx.x * blockDim.x + threadIdx.x;   // nRows*32 float4s
    if (i >= nQuads) return;
    float r = 1.0f / fmaxf(deg[i >> 5], 1.0f);       // 32 quads per row
    float4 v = sum4[i];
    v.x *= r; v.y *= r; v.z *= r; v.w *= r;
    sum4[i] = v;
}

// ---------------------------------------------------------------------------
// GEMM helpers
// ---------------------------------------------------------------------------
__device__ __forceinline__ void hgcn_stage(const float* __restrict__ g,
                                           float* l, int tid) {
#if HGCN_ASYNC_LDS
    // Builtin wants non-const pointers: (global src, lds dst, imm off, imm cpol)
    v4i __attribute__((address_space(1)))* gs =
        (v4i __attribute__((address_space(1)))*)g;
    v4i __attribute__((address_space(3)))* ld =
        (v4i __attribute__((address_space(3)))*)l;
    __builtin_amdgcn_global_load_async_to_lds_b128(gs + tid,       ld + tid,       0, 0);
    __builtin_amdgcn_global_load_async_to_lds_b128(gs + tid + 256, ld + tid + 256, 0, 0);
#else
    const float4* gs = (const float4*)g;
    float4* ld = (float4*)l;
    ld[tid]       = gs[tid];
    ld[tid + 256] = gs[tid + 256];
#endif
}

// One accumulation term: acc += A_strip(16xK from LDS) @ WT^T tile.
// WT is N-major (transposed weights), so the B fragment {W[k][n], W[k+1][n]}
// is one contiguous float2 at WT[n*DD + k].
__device__ __forceinline__ v8f hgcn_term(v8f acc, const float* lA,
                                         const float* __restrict__ WT,
                                         int m, int kb, int col0) {
    const float*  Arow = lA + m * DD;
    const float2* wrow = (const float2*)(WT + (size_t)(col0 + m) * DD);
    #pragma unroll
    for (int k0 = 0; k0 < DD; k0 += 4) {
        v2f a, b;
        a.x = Arow[k0 + kb];
        a.y = Arow[k0 + kb + 1];
        float2 w = wrow[(k0 + kb) >> 1];
        b.x = w.x;
        b.y = w.y;
        acc = __builtin_amdgcn_wmma_f32_16x16x4_f32(
            false, a, false, b, (short)0, acc, false, false);
    }
    return acc;
}

// ---------------------------------------------------------------------------
// Kernel 4: fused multi-term GEMM via V_WMMA_F32_16X16X4_F32.
// Block = 256 threads = 8 waves; block owns rows [row0,row0+16) x all 128
// cols; wave w computes the 16x16 tile at cols [16w,16w+16).
// Fragment layouts per CDNA5 ISA 7.12.2 (f32, wave32):
//   A 16x4 : lane m=lane&15 holds A[m][kb],A[m][kb+1], kb=2*(lane>>4)
//   B 4x16 : lane holds B[kb][n],B[kb+1][n], n=lane&15
//   C/D    : VGPR j holds row M=j+8*(lane>>4), col N=lane&15
// ---------------------------------------------------------------------------
template <int TERMS>
__global__ __launch_bounds__(256)
void hgcn_gemm_kernel(const float* __restrict__ A0, const float* __restrict__ WT0,
                      const float* __restrict__ A1, const float* __restrict__ WT1,
                      const float* __restrict__ A2, const float* __restrict__ WT2,
                      const float* __restrict__ b0, const float* __restrict__ b1,
                      float* __restrict__ out) {
    __shared__ float lA[3 * 16 * DD];                // up to 3 strips, 8 KB each

    const int row0 = blockIdx.x * 16;
    const int tid  = threadIdx.x;

    hgcn_stage(A0 + (size_t)row0 * DD, &lA[0 * 16 * DD], tid);
    hgcn_stage(A1 + (size_t)row0 * DD, &lA[1 * 16 * DD], tid);
    if (TERMS > 2) hgcn_stage(A2 + (size_t)row0 * DD, &lA[2 * 16 * DD], tid);
#if HGCN_ASYNC_LDS
    HGCN_WAIT_ASYNC();
#endif
    __syncthreads();

    const int lane = tid & 31;
    const int col0 = (tid >> 5) * 16;
    const int m    = lane & 15;
    const int kb   = (lane >> 4) * 2;

    v8f acc = {};
    acc = hgcn_term(acc, &lA[0 * 16 * DD], WT0, m, kb, col0);
    acc = hgcn_term(acc, &lA[1 * 16 * DD], WT1, m, kb, col0);
    if (TERMS > 2) acc = hgcn_term(acc, &lA[2 * 16 * DD], WT2, m, kb, col0);

    float bias = b0[col0 + m];
    if (TERMS > 2) bias += b1[col0 + m];

    const int nbase = col0 + m;
    const int mbase = (lane >> 4) * 8;
    #pragma unroll
    for (int j = 0; j < 8; ++j) {
        out[(size_t)(row0 + mbase + j) * DD + nbase] = acc[j] + bias;
    }
}

// ---------------------------------------------------------------------------
// Launch
// ---------------------------------------------------------------------------
extern "C" void kernel_launch(void* const* d_in, const int* in_sizes, int n_in,
                              void* d_out, int out_size, void* d_ws, size_t ws_size,
                              hipStream_t stream) {
    (void)in_sizes; (void)n_in; (void)out_size; (void)ws_size;

    const float* x_gene  = (const float*)d_in[0];
    const float* x_trait = (const float*)d_in[1];
    const int*   src_gg  = (const int*)d_in[2];
    const int*   dst_gg  = (const int*)d_in[3];
    const int*   src_gt  = (const int*)d_in[4];
    const int*   dst_gt  = (const int*)d_in[5];
    const int*   src_tg  = (const int*)d_in[6];
    const int*   dst_tg  = (const int*)d_in[7];
    const float* Wn_gg   = (const float*)d_in[8];
    const float* Ws_gg   = (const float*)d_in[9];
    const float* b_gg    = (const float*)d_in[10];
    const float* Wn_gt   = (const float*)d_in[11];
    const float* Ws_gt   = (const float*)d_in[12];
    const float* b_gt    = (const float*)d_in[13];
    const float* Wn_tg   = (const float*)d_in[14];
    const float* Ws_tg   = (const float*)d_in[15];
    const float* b_tg    = (const float*)d_in[16];

    float* h_gene  = (float*)d_out;                       // [NG, 128]
    float* h_trait = (float*)d_out + (size_t)NG * DD;     // [NT, 128]

    // Workspace layout
    float* ws = (float*)d_ws;
    const size_t SUM = (size_t)NG * DD;                   // 12.8M floats
    float* sum_gg = ws;
    float* sum_gt = ws + SUM;
    float* sum_tg = ws + 2 * SUM;
    float* deg_gg = ws + 3 * SUM;
    float* deg_gt = deg_gg + NG;
    float* deg_tg = deg_gt + NT;
    float* WT     = deg_tg + NG;                          // 5 x 128*128

    // Zero sums + degrees (graph-capturable memset node).
    (void)hipMemsetAsync(d_ws, 0,
                         (3 * SUM + (size_t)(NG + NT + NG)) * sizeof(float),
                         stream);

    // Transposed / combined weights.
    hgcn_wprep_kernel<<<(DD * DD + 255) / 256, 256, 0, stream>>>(
        Ws_gg, Ws_tg, Wn_gg, Wn_tg, Ws_gt, Wn_gt, WT);

    // Edge scatters: one wave per edge, 8 waves per block.
    const int sblocks = (NE * 32 + 255) / 256;            // 62500
    hgcn_scatter_kernel<<<sblocks, 256, 0, stream>>>(x_gene,  src_gg, dst_gg,
                                                     sum_gg, deg_gg, NE);
    hgcn_scatter_kernel<<<sblocks, 256, 0, stream>>>(x_gene,  src_gt, dst_gt,
                                                     sum_gt, deg_gt, NE);
    hgcn_scatter_kernel<<<sblocks, 256, 0, stream>>>(x_trait, src_tg, dst_tg,
                                                     sum_tg, deg_tg, NE);

    // Mean normalization (float4-wide).
    const int nQuads  = (int)(SUM / 4);                   // 3.2M per buffer
    const int mblocks = (nQuads + 255) / 256;
    hgcn_mean_kernel<<<mblocks, 256, 0, stream>>>((float4*)sum_gg, deg_gg, nQuads);
    hgcn_mean_kernel<<<mblocks, 256, 0, stream>>>((float4*)sum_gt, deg_gt, nQuads);
    hgcn_mean_kernel<<<mblocks, 256, 0, stream>>>((float4*)sum_tg, deg_tg, nQuads);

    // Fused WMMA GEMMs.
    // gene:  x_gene@(Ws_gg+Ws_tg) + hn_gg@Wn_gg + hn_tg@Wn_tg + b_gg + b_tg
    hgcn_gemm_kernel<3><<<NG / 16, 256, 0, stream>>>(
        x_gene, WT + 0 * DD * DD,
        sum_gg, WT + 1 * DD * DD,
        sum_tg, WT + 2 * DD * DD,
        b_gg, b_tg, h_gene);
    // trait: x_trait@Ws_gt + hn_gt@Wn_gt + b_gt
    hgcn_gemm_kernel<2><<<NT / 16, 256, 0, stream>>>(
        x_trait, WT + 3 * DD * DD,
        sum_gt,  WT + 4 * DD * DD,
        (const float*)nullptr, (const float*)nullptr,
        b_gt, (const float*)nullptr, h_trait);
}